// XGGuard_4483945857491
// MI455X (gfx1250) — compile-verified
//
#include <hip/hip_runtime.h>
#include <cstdint>

#define Nn 32768
#define Gg 256
#define Ee 524288
#define Tt 16
#define FSc 256
#define FTc 128

typedef float v2f __attribute__((ext_vector_type(2)));
typedef float v8f __attribute__((ext_vector_type(8)));

// ---------- utility kernels ----------
__global__ __launch_bounds__(256) void k_fill(float* __restrict__ p, float v, int n) {
  int i = blockIdx.x * 256 + threadIdx.x;
  if (i < n) p[i] = v;
}

// in-degree histogram (int)
__global__ __launch_bounds__(256)
void k_count_int(const int* __restrict__ ei, int* __restrict__ cnt, int E) {
  int e = blockIdx.x * 256 + threadIdx.x;
  if (e < E) atomicAdd(&cnt[ei[E + e]], 1);     // dst row of edge_index[2,E]
}

// dinv[n] = 1/sqrt(cnt[n] + 1)  (self-loop)
__global__ __launch_bounds__(256)
void k_dinv_from_cnt(const int* __restrict__ cnt, float* __restrict__ dinv, int n) {
  int i = blockIdx.x * 256 + threadIdx.x;
  if (i < n) dinv[i] = 1.0f / sqrtf((float)cnt[i] + 1.0f);
}

// single-block exclusive scan over n (n multiple of 256): rowstart[0..n], cursor copy
__global__ __launch_bounds__(256)
void k_scan(const int* __restrict__ cnt, int* __restrict__ rowstart,
            int* __restrict__ cursor, int n) {
  __shared__ int s[256];
  int tid = threadIdx.x;
  int chunk = n / 256;
  int base = tid * chunk;
  int sum = 0;
  for (int i = 0; i < chunk; ++i) sum += cnt[base + i];
  s[tid] = sum;
  __syncthreads();
  for (int off = 1; off < 256; off <<= 1) {
    int v = (tid >= off) ? s[tid - off] : 0;
    __syncthreads();
    s[tid] += v;
    __syncthreads();
  }
  int prefix = (tid == 0) ? 0 : s[tid - 1];
  for (int i = 0; i < chunk; ++i) {
    rowstart[base + i] = prefix;
    cursor[base + i] = prefix;
    prefix += cnt[base + i];
  }
  if (tid == 255) rowstart[n] = prefix;
}

// scatter edge sources into CSR slots
__global__ __launch_bounds__(256)
void k_scatter(const int* __restrict__ ei, int* __restrict__ cursor,
               int* __restrict__ csr_src, int E) {
  int e = blockIdx.x * 256 + threadIdx.x;
  if (e >= E) return;
  int src = ei[e];
  int dst = ei[E + e];
  int pos = atomicAdd(&cursor[dst], 1);
  csr_src[pos] = src;
}

// ---------- GEMM: C[M,Nout] = X[M,K] @ W[K,Nout], f32 WMMA 16x16x4 ----------
// One wave -> 16x64 strip (4 accumulators). K multiple of 4, M mult of 16, Nout mult of 64.
__global__ __launch_bounds__(256)
void k_gemm_f32_wmma(const float* __restrict__ X, const float* __restrict__ W,
                     float* __restrict__ C, int M, int K, int Nout) {
  const int lane  = threadIdx.x & 31;
  const int wave  = blockIdx.x * 8 + (threadIdx.x >> 5);
  const int tilesN = Nout >> 6;
  const int tm = wave / tilesN;
  const int tn = wave - tm * tilesN;
  if (tm * 16 >= M) return;                 // wave-uniform: EXEC stays all-1 for WMMA
  const int m0 = tm * 16;
  const int n0 = tn * 64;
  const int lmod  = lane & 15;
  const int khalf = (lane >> 4) << 1;       // 0 or 2 (K pair held by this lane half)

  v8f acc0 = {}, acc1 = {}, acc2 = {}, acc3 = {};
  const float* __restrict__ xrow = X + (size_t)(m0 + lmod) * K + khalf;
  for (int k = 0; k < K; k += 4) {
    v2f a;
    a.x = xrow[k];
    a.y = xrow[k + 1];
    const float* __restrict__ wb = W + (size_t)(k + khalf) * Nout + n0 + lmod;
    v2f b0, b1, b2, b3;
    b0.x = wb[0];   b0.y = wb[Nout +  0];
    b1.x = wb[16];  b1.y = wb[Nout + 16];
    b2.x = wb[32];  b2.y = wb[Nout + 32];
    b3.x = wb[48];  b3.y = wb[Nout + 48];
    acc0 = __builtin_amdgcn_wmma_f32_16x16x4_f32(false, a, false, b0, (short)0, acc0, false, false);
    acc1 = __builtin_amdgcn_wmma_f32_16x16x4_f32(false, a, false, b1, (short)0, acc1, false, false);
    acc2 = __builtin_amdgcn_wmma_f32_16x16x4_f32(false, a, false, b2, (short)0, acc2, false, false);
    acc3 = __builtin_amdgcn_wmma_f32_16x16x4_f32(false, a, false, b3, (short)0, acc3, false, false);
  }
  const int rbase = m0 + ((lane >> 4) << 3);   // lanes 16-31 hold rows M+8..M+15
  float* __restrict__ cb = C + (size_t)rbase * Nout + n0 + lmod;
  #pragma unroll
  for (int r = 0; r < 8; ++r) {
    cb[(size_t)r * Nout +  0] = acc0[r];
    cb[(size_t)r * Nout + 16] = acc1[r];
    cb[(size_t)r * Nout + 32] = acc2[r];
    cb[(size_t)r * Nout + 48] = acc3[r];
  }
}

// ---------- fused GCN gather + combine (atomic-free) ----------
// One wave per destination node:
//   out[node] = sum_{e in CSR(node)} h[src_e]*dinv[src_e]*dinv[node]
//             + h[node]*dinv[node]^2 + bias (+resid) (+relu)
template<int F>
__global__ __launch_bounds__(256)
void k_gcn_gather(const int* __restrict__ rowstart, const int* __restrict__ csr_src,
                  const float* __restrict__ h, const float* __restrict__ dinv,
                  const float* __restrict__ bias, const float* __restrict__ resid,
                  float* __restrict__ out, int n, int do_relu) {
  constexpr int J = F / 128;                 // float4 chunks per lane
  int node = blockIdx.x * 8 + (threadIdx.x >> 5);
  if (node >= n) return;
  int lane = threadIdx.x & 31;
  int beg = rowstart[node], end = rowstart[node + 1];
  float dn = dinv[node];

  float4 acc[J];
  #pragma unroll
  for (int j = 0; j < J; ++j) acc[j] = make_float4(0.f, 0.f, 0.f, 0.f);

  for (int e = beg; e < end; ++e) {
    int src = csr_src[e];
    float norm = dinv[src] * dn;
    const float4* __restrict__ hv = (const float4*)(h + (size_t)src * F);
    #pragma unroll
    for (int j = 0; j < J; ++j) {
      float4 v = hv[lane + 32 * j];
      acc[j].x += v.x * norm;
      acc[j].y += v.y * norm;
      acc[j].z += v.z * norm;
      acc[j].w += v.w * norm;
    }
  }

  const float4* __restrict__ hd = (const float4*)(h + (size_t)node * F);
  const float4* __restrict__ bv = (const float4*)bias;
  const float4* __restrict__ rv = resid ? (const float4*)(resid + (size_t)node * F) : nullptr;
  float4* __restrict__ ov = (float4*)(out + (size_t)node * F);
  float d2 = dn * dn;
  #pragma unroll
  for (int j = 0; j < J; ++j) {
    int idx = lane + 32 * j;
    float4 hs4 = hd[idx];
    float4 bb = bv[idx];
    float4 v;
    v.x = acc[j].x + hs4.x * d2 + bb.x;
    v.y = acc[j].y + hs4.y * d2 + bb.y;
    v.z = acc[j].z + hs4.z * d2 + bb.z;
    v.w = acc[j].w + hs4.w * d2 + bb.w;
    if (rv) {
      float4 rr = rv[idx];
      v.x += rr.x; v.y += rr.y; v.z += rr.z; v.w += rr.w;
    }
    if (do_relu) {
      v.x = fmaxf(v.x, 0.f); v.y = fmaxf(v.y, 0.f);
      v.z = fmaxf(v.z, 0.f); v.w = fmaxf(v.w, 0.f);
    }
    ov[idx] = v;
  }
}

// ---------- segment means ----------
__global__ __launch_bounds__(256)
void k_counts(const int* __restrict__ batch, float* __restrict__ counts, int n) {
  int i = blockIdx.x * 256 + threadIdx.x;
  if (i < n) atomicAdd(&counts[batch[i]], 1.0f);
}

__global__ __launch_bounds__(256)
void k_seg_acc(const float* __restrict__ x, const int* __restrict__ batch,
               float* __restrict__ p, int total, int logF) {
  int i = blockIdx.x * 256 + threadIdx.x;
  if (i >= total) return;
  int node = i >> logF;
  int f = i & ((1 << logF) - 1);
  atomicAdd(&p[((size_t)batch[node] << logF) + f], x[i]);
}

__global__ __launch_bounds__(256)
void k_div_counts(float* __restrict__ p, const float* __restrict__ counts, int total, int logF) {
  int i = blockIdx.x * 256 + threadIdx.x;
  if (i < total) p[i] /= counts[i >> logF];
}

// ---------- mean_agent = h_t + tokens.mean(axis=1), float4 vectorized ----------
__global__ __launch_bounds__(256)
void k_mean_agent(const float* __restrict__ ht, const float* __restrict__ tokens,
                  float* __restrict__ ma, int total4) {   // total4 = Nn * 32
  int i = blockIdx.x * 256 + threadIdx.x;
  if (i >= total4) return;
  int node = i >> 5;                 // 32 float4 per node (FT=128)
  int q = i & 31;
  const float4* __restrict__ tp = (const float4*)(tokens + (size_t)node * (Tt * FTc)) + q;
  float4 s = make_float4(0.f, 0.f, 0.f, 0.f);
  #pragma unroll
  for (int t = 0; t < Tt; ++t) {
    float4 v = tp[t * (FTc / 4)];
    s.x += v.x; s.y += v.y; s.z += v.z; s.w += v.w;
  }
  float4 hv = ((const float4*)ht)[i];
  float4 r;
  r.x = hv.x + s.x * (1.0f / Tt);
  r.y = hv.y + s.y * (1.0f / Tt);
  r.z = hv.z + s.z * (1.0f / Tt);
  r.w = hv.w + s.w * (1.0f / Tt);
  ((float4*)ma)[i] = r;
}

__device__ __forceinline__ float waveReduceSum(float v) {
  #pragma unroll
  for (int m = 16; m >= 1; m >>= 1) v += __shfl_xor(v, m, 32);
  return v;
}

// ---------- per-node pos/neg dot-product scores (one wave per node) ----------
__global__ __launch_bounds__(256)
void k_scores(const float* __restrict__ hs, const float* __restrict__ ma,
              const float* __restrict__ ps, const float* __restrict__ pt,
              const int* __restrict__ batch, const int* __restrict__ nbatch,
              float* __restrict__ ssp, float* __restrict__ stp,
              float* __restrict__ ssn, float* __restrict__ stn, int n) {
  int node = blockIdx.x * 8 + (threadIdx.x >> 5);
  if (node >= n) return;
  int lane = threadIdx.x & 31;
  int gp = batch[node], gn = nbatch[node];
  const float* hrow = hs + (size_t)node * FSc;
  const float* pp = ps + (size_t)gp * FSc;
  const float* pn = ps + (size_t)gn * FSc;
  float ap = 0.0f, an = 0.0f;
  #pragma unroll
  for (int f = lane; f < FSc; f += 32) { float v = hrow[f]; ap += v * pp[f]; an += v * pn[f]; }
  const float* mrow = ma + (size_t)node * FTc;
  const float* qp = pt + (size_t)gp * FTc;
  const float* qn = pt + (size_t)gn * FTc;
  float bp = 0.0f, bn = 0.0f;
  #pragma unroll
  for (int f = lane; f < FTc; f += 32) { float v = mrow[f]; bp += v * qp[f]; bn += v * qn[f]; }
  ap = waveReduceSum(ap); an = waveReduceSum(an);
  bp = waveReduceSum(bp); bn = waveReduceSum(bn);
  if (lane == 0) { ssp[node] = ap; stp[node] = bp; ssn[node] = an; stn[node] = bn; }
}

// ---------- global moments for _fusion: sum_a, sum_a2, sum_b, sum_b2, sum_ab ----------
__global__ __launch_bounds__(256)
void k_stats(const float* __restrict__ a, const float* __restrict__ b,
             float* __restrict__ acc, int n) {
  int tid = blockIdx.x * 256 + threadIdx.x;
  int stride = gridDim.x * 256;
  float sa = 0, sa2 = 0, sb = 0, sb2 = 0, sab = 0;
  for (int i = tid; i < n; i += stride) {
    float va = a[i], vb = b[i];
    sa += va; sa2 += va * va; sb += vb; sb2 += vb * vb; sab += va * vb;
  }
  sa = waveReduceSum(sa); sa2 = waveReduceSum(sa2);
  sb = waveReduceSum(sb); sb2 = waveReduceSum(sb2);
  sab = waveReduceSum(sab);
  if ((threadIdx.x & 31) == 0) {
    atomicAdd(acc + 0, sa);  atomicAdd(acc + 1, sa2);
    atomicAdd(acc + 2, sb);  atomicAdd(acc + 3, sb2);
    atomicAdd(acc + 4, sab);
  }
}

// ---------- final fusion: sigmoid(zs + cov*zt), pos -> out[0:N), neg -> out[N:2N) ----------
__global__ __launch_bounds__(256)
void k_fusion(const float* __restrict__ ssp, const float* __restrict__ stp,
              const float* __restrict__ ssn, const float* __restrict__ stn,
              const float* __restrict__ acc, float* __restrict__ out, int n) {
  int i = blockIdx.x * 256 + threadIdx.x;
  if (i >= n) return;
  float fn = (float)n;
  {
    float ms = acc[0] / fn, mt = acc[2] / fn;
    float vs = (acc[1] - fn * ms * ms) / (fn - 1.0f);
    float vt = (acc[3] - fn * mt * mt) / (fn - 1.0f);
    float ds = sqrtf(fmaxf(vs, 0.0f)) + 1e-10f;
    float dt = sqrtf(fmaxf(vt, 0.0f)) + 1e-10f;
    float cov = (acc[4] - fn * ms * mt) / (fn * ds * dt);
    float x = (ssp[i] - ms) / ds + cov * ((stp[i] - mt) / dt);
    out[i] = 1.0f / (1.0f + expf(-x));
  }
  {
    float ms = acc[5] / fn, mt = acc[7] / fn;
    float vs = (acc[6] - fn * ms * ms) / (fn - 1.0f);
    float vt = (acc[8] - fn * mt * mt) / (fn - 1.0f);
    float ds = sqrtf(fmaxf(vs, 0.0f)) + 1e-10f;
    float dt = sqrtf(fmaxf(vt, 0.0f)) + 1e-10f;
    float cov = (acc[9] - fn * ms * mt) / (fn * ds * dt);
    float x = (ssn[i] - ms) / ds + cov * ((stn[i] - mt) / dt);
    out[n + i] = 1.0f / (1.0f + expf(-x));
  }
}

extern "C" void kernel_launch(void* const* d_in, const int* in_sizes, int n_in,
                              void* d_out, int out_size, void* d_ws, size_t ws_size,
                              hipStream_t stream) {
  (void)in_sizes; (void)n_in; (void)out_size; (void)ws_size;
  const float* x_s   = (const float*)d_in[0];
  const float* x_t   = (const float*)d_in[1];
  const float* toks  = (const float*)d_in[2];
  const int*   ei_s  = (const int*)d_in[3];
  const int*   ei_t  = (const int*)d_in[4];
  const int*   batch = (const int*)d_in[5];
  const int*   nbat  = (const int*)d_in[6];
  const float* W_s1  = (const float*)d_in[7];
  const float* b_s1  = (const float*)d_in[8];
  const float* W_s2  = (const float*)d_in[9];
  const float* b_s2  = (const float*)d_in[10];
  const float* W_t1  = (const float*)d_in[11];
  const float* b_t1  = (const float*)d_in[12];
  const float* W_t2  = (const float*)d_in[13];
  const float* b_t2  = (const float*)d_in[14];
  float* out = (float*)d_out;

  float* ws = (float*)d_ws;
  size_t off = 0;
  float* buf0   = ws + off; off += (size_t)Nn * 256;   // GEMM outputs / mean_agent
  float* buf1   = ws + off; off += (size_t)Nn * 256;   // h1_t / h_t
  float* hs     = ws + off; off += (size_t)Nn * 256;   // h1_s then final h_s
  float* dinv_s = ws + off; off += Nn;
  float* dinv_t = ws + off; off += Nn;
  float* ps     = ws + off; off += (size_t)Gg * 256;
  float* pt     = ws + off; off += (size_t)Gg * 128;
  float* counts = ws + off; off += Gg;
  float* ssp    = ws + off; off += Nn;
  float* stp    = ws + off; off += Nn;
  float* ssn    = ws + off; off += Nn;
  float* stn    = ws + off; off += Nn;
  float* acc    = ws + off; off += 16;
  // CSR structures (int)
  int* cnt_s = (int*)(ws + off); off += Nn;
  int* cnt_t = (int*)(ws + off); off += Nn;
  int* rs_s  = (int*)(ws + off); off += Nn + 1;
  int* rs_t  = (int*)(ws + off); off += Nn + 1;
  int* cur_s = (int*)(ws + off); off += Nn;
  int* cur_t = (int*)(ws + off); off += Nn;
  int* csr_s = (int*)(ws + off); off += Ee;
  int* csr_t = (int*)(ws + off); off += Ee;

  dim3 blk(256);
  auto cdiv = [](int a, int b) { return (a + b - 1) / b; };
  const int NF128 = Nn * 128;
  const int gemmBlocks256 = (Nn / 16) * (256 / 64) / 8;   // 1024
  const int gemmBlocks128 = (Nn / 16) * (128 / 64) / 8;   // 512

  // ---- build CSR for both graphs + dinv
  k_fill<<<cdiv(Nn, 256), blk, 0, stream>>>((float*)cnt_s, 0.0f, Nn);   // bitwise zero
  k_fill<<<cdiv(Nn, 256), blk, 0, stream>>>((float*)cnt_t, 0.0f, Nn);
  k_count_int<<<cdiv(Ee, 256), blk, 0, stream>>>(ei_s, cnt_s, Ee);
  k_count_int<<<cdiv(Ee, 256), blk, 0, stream>>>(ei_t, cnt_t, Ee);
  k_dinv_from_cnt<<<cdiv(Nn, 256), blk, 0, stream>>>(cnt_s, dinv_s, Nn);
  k_dinv_from_cnt<<<cdiv(Nn, 256), blk, 0, stream>>>(cnt_t, dinv_t, Nn);
  k_scan<<<1, blk, 0, stream>>>(cnt_s, rs_s, cur_s, Nn);
  k_scan<<<1, blk, 0, stream>>>(cnt_t, rs_t, cur_t, Nn);
  k_scatter<<<cdiv(Ee, 256), blk, 0, stream>>>(ei_s, cur_s, csr_s, Ee);
  k_scatter<<<cdiv(Ee, 256), blk, 0, stream>>>(ei_t, cur_t, csr_t, Ee);

  // ---- s layer 1: h1_s = relu(gcn(x_s, W_s1)) -> hs
  k_gemm_f32_wmma<<<gemmBlocks256, blk, 0, stream>>>(x_s, W_s1, buf0, Nn, 256, 256);
  k_gcn_gather<256><<<Nn / 8, blk, 0, stream>>>(rs_s, csr_s, buf0, dinv_s, b_s1, nullptr, hs, Nn, 1);

  // ---- s layer 2: h_s = gcn(h1_s, W_s2) + x_s -> hs
  k_gemm_f32_wmma<<<gemmBlocks256, blk, 0, stream>>>(hs, W_s2, buf0, Nn, 256, 256);
  k_gcn_gather<256><<<Nn / 8, blk, 0, stream>>>(rs_s, csr_s, buf0, dinv_s, b_s2, x_s, hs, Nn, 0);

  // ---- t layer 1: h1_t = relu(gcn(x_t, W_t1)) -> buf1
  k_gemm_f32_wmma<<<gemmBlocks256, blk, 0, stream>>>(x_t, W_t1, buf0, Nn, 128, 256);
  k_gcn_gather<256><<<Nn / 8, blk, 0, stream>>>(rs_t, csr_t, buf0, dinv_t, b_t1, nullptr, buf1, Nn, 1);

  // ---- t layer 2: h_t = gcn(h1_t, W_t2) -> buf1[0:N*128]
  k_gemm_f32_wmma<<<gemmBlocks128, blk, 0, stream>>>(buf1, W_t2, buf0, Nn, 256, 128);
  k_gcn_gather<128><<<Nn / 8, blk, 0, stream>>>(rs_t, csr_t, buf0, dinv_t, b_t2, nullptr, buf1, Nn, 0);

  // mean_agent = h_t + tokens.mean(1) -> buf0[0:N*128]
  k_mean_agent<<<cdiv(Nn * 32, 256), blk, 0, stream>>>(buf1, toks, buf0, Nn * 32);

  // prototypes
  k_fill<<<1, blk, 0, stream>>>(counts, 0.0f, Gg);
  k_counts<<<cdiv(Nn, 256), blk, 0, stream>>>(batch, counts, Nn);
  k_fill<<<cdiv(Gg * 256, 256), blk, 0, stream>>>(ps, 0.0f, Gg * 256);
  k_seg_acc<<<cdiv(Nn * 256, 256), blk, 0, stream>>>(hs, batch, ps, Nn * 256, 8);
  k_div_counts<<<cdiv(Gg * 256, 256), blk, 0, stream>>>(ps, counts, Gg * 256, 8);
  k_fill<<<cdiv(Gg * 128, 256), blk, 0, stream>>>(pt, 0.0f, Gg * 128);
  k_seg_acc<<<cdiv(NF128, 256), blk, 0, stream>>>(buf0, batch, pt, NF128, 7);
  k_div_counts<<<cdiv(Gg * 128, 256), blk, 0, stream>>>(pt, counts, Gg * 128, 7);

  // scores + fusion
  k_scores<<<Nn / 8, blk, 0, stream>>>(hs, buf0, ps, pt, batch, nbat, ssp, stp, ssn, stn, Nn);
  k_fill<<<1, blk, 0, stream>>>(acc, 0.0f, 16);
  k_stats<<<256, blk, 0, stream>>>(ssp, stp, acc, Nn);
  k_stats<<<256, blk, 0, stream>>>(ssn, stn, acc + 5, Nn);
  k_fusion<<<cdiv(Nn, 256), blk, 0, stream>>>(ssp, stp, ssn, stn, acc, out, Nn);
}